// CustomDistll_19705309954434
// MI455X (gfx1250) — compile-verified
//
#include <hip/hip_runtime.h>
#include <stdint.h>

typedef __attribute__((ext_vector_type(2))) float v2f;
typedef __attribute__((ext_vector_type(4))) float v4f;
typedef __attribute__((ext_vector_type(8))) float v8f;

#define NBINS 64
#define C_    256
#define N_    4
#define H_    128
#define W_    160
#define HW_   (H_ * W_)      // 20480
#define P_    (N_ * HW_)     // 81920
#define EPS_  1e-12f

// ---- workspace layout (byte offsets into d_ws) ----
#define OFF_IDX    0                         // P_ bytes of uint8 bin ids
#define OFF_SUMS_S 81920                     // 65*256 f32
#define OFF_SUMS_T (OFF_SUMS_S + 66560)      // 65*256 f32
#define OFF_CNT    (OFF_SUMS_T + 66560)      // 65 f32 (contiguous with sums for zeroing)
#define OFF_PS     (OFF_CNT + 512)           // 64*256 f32
#define OFF_PT     (OFF_PS + 65536)          // 64*256 f32
#define ZERO_FLOATS ((OFF_CNT + 65 * 4 - OFF_SUMS_S) / 4)   // sumsS+sumsT+counts = 33345

// Zero the accumulator region (sumsS, sumsT, counts are contiguous) and d_out.
__global__ void k_zero(float* acc, float* out) {
    int i = blockIdx.x * blockDim.x + threadIdx.x;
    if (i < ZERO_FLOATS) acc[i] = 0.0f;
    if (i == 0) out[0] = 0.0f;
}

__device__ __forceinline__ int bin_of(float d) {
    float f = d * (float)NBINS;                    // (d - 0) / (1/64)
    int b;
    // !(f >= 0) catches NaN and negatives; f > 64 catches overflow and +inf.
    if (!(f >= 0.0f) || f > (float)NBINS) b = NBINS;
    else                                  b = (int)f;   // trunc toward zero
    if (b > NBINS) b = NBINS;
    return b;
}

// Uniform-discretization bin index per pixel + global counts per bin.
// Vectorized: each thread handles 4 consecutive pixels (P_ % (256*4) == 0).
__global__ void k_idx(const float* __restrict__ depth, uint8_t* __restrict__ idx,
                      float* __restrict__ counts) {
    __shared__ float lcnt[NBINS + 1];
    int t = threadIdx.x;
    if (t <= NBINS) lcnt[t] = 0.0f;
    __syncthreads();
    int p4 = (blockIdx.x * blockDim.x + t) * 4;
    if (p4 < P_) {
        v4f d = *(const v4f*)(depth + p4);
        uint32_t packed = 0;
#pragma unroll
        for (int k = 0; k < 4; ++k) {
            int b = bin_of(d[k]);
            packed |= ((uint32_t)b) << (8 * k);
            atomicAdd(&lcnt[b], 1.0f);
        }
        *(uint32_t*)(idx + p4) = packed;           // 4 bin ids in one b32 store
    }
    __syncthreads();
    if (t <= NBINS) atomicAdd(&counts[t], lcnt[t]);
}

// One block per (tensor, n, c) plane: stream 20480 floats as b128 loads,
// scatter into a 65-entry LDS histogram with ds_add_f32.
__global__ void k_accum(const float* __restrict__ predsS, const float* __restrict__ predsT,
                        const uint8_t* __restrict__ idx,
                        float* __restrict__ sumsS, float* __restrict__ sumsT) {
    __shared__ float lbins[NBINS + 1];
    int bid   = blockIdx.x;
    int tsel  = bid / (N_ * C_);          // 0 = S, 1 = T
    int plane = bid % (N_ * C_);          // = n*C_ + c (matches NCHW plane order)
    int n     = plane / C_;
    int c     = plane % C_;

    const float*   src  = (tsel == 0 ? predsS : predsT) + (size_t)plane * HW_;
    const uint8_t* bidx = idx + (size_t)n * HW_;
    float*         sums = (tsel == 0 ? sumsS : sumsT);

    int tid = threadIdx.x;
    if (tid <= NBINS) lbins[tid] = 0.0f;
    __syncthreads();

    // HW_ / 4 = 5120 float4 elements, 256 threads -> 20 iterations.
    for (int q = tid; q < HW_ / 4; q += blockDim.x) {
        v4f v = *(const v4f*)(src + q * 4);                 // global_load_b128
        uint32_t b4 = *(const uint32_t*)(bidx + q * 4);     // 4 packed bin ids
#pragma unroll
        for (int k = 0; k < 4; ++k)
            atomicAdd(&lbins[(b4 >> (8 * k)) & 0xFF], v[k]);
    }

    __syncthreads();
    if (tid <= NBINS) atomicAdd(&sums[tid * C_ + c], lbins[tid]);
}

// One block (256 threads) per bin row: mean -> L2 normalize -> prototype row.
__global__ void k_protos(const float* __restrict__ sums, const float* __restrict__ counts,
                         float* __restrict__ protos) {
    __shared__ float red[C_];
    int r = blockIdx.x;       // 0..63
    int c = threadIdx.x;      // 0..255
    float cnt = counts[r];
    float m = sums[r * C_ + c] / fmaxf(cnt, 1.0f);
    red[c] = m * m;
    __syncthreads();
    for (int s = C_ / 2; s > 0; s >>= 1) {
        if (c < s) red[c] += red[c + s];
        __syncthreads();
    }
    float scale = 1.0f / fmaxf(sqrtf(red[0]), EPS_);
    protos[r * C_ + c] = m * scale;
}

// One wave32 per 16x16 tile of the 64x64 Gram matrices.
// sim = P @ P^T via V_WMMA_F32_16X16X4_F32; accumulate mean((simS - simT)^2).
__global__ void __launch_bounds__(32) k_gram(const float* __restrict__ pS,
                                             const float* __restrict__ pT,
                                             float* __restrict__ out) {
    int tile = blockIdx.x;                 // 0..15
    int i0 = (tile >> 2) * 16;
    int j0 = (tile & 3) * 16;
    int L  = threadIdx.x;                  // 0..31, EXEC all ones (no divergence)
    int m  = L & 15;                       // ISA 16x4 f32 A layout: lane -> M = L%16
    int kh = (L >> 4) * 2;                 // lanes 16-31 hold K = 2,3

    v8f cs = {};
    v8f ct = {};
    const float* aS = pS + (size_t)(i0 + m) * C_;
    const float* bS = pS + (size_t)(j0 + m) * C_;   // Gram: B[k][n] = P[j0+n][k], same lane formula
    const float* aT = pT + (size_t)(i0 + m) * C_;
    const float* bT = pT + (size_t)(j0 + m) * C_;

    for (int k0 = 0; k0 < C_; k0 += 4) {
        v2f a_s = *(const v2f*)(aS + k0 + kh);
        v2f b_s = *(const v2f*)(bS + k0 + kh);
        v2f a_t = *(const v2f*)(aT + k0 + kh);
        v2f b_t = *(const v2f*)(bT + k0 + kh);
        cs = __builtin_amdgcn_wmma_f32_16x16x4_f32(false, a_s, false, b_s,
                                                   (short)0, cs, false, false);
        ct = __builtin_amdgcn_wmma_f32_16x16x4_f32(false, a_t, false, b_t,
                                                   (short)0, ct, false, false);
    }

    float acc = 0.0f;
#pragma unroll
    for (int v = 0; v < 8; ++v) {
        float d = cs[v] - ct[v];
        acc += d * d;
    }
    for (int off = 16; off > 0; off >>= 1)
        acc += __shfl_xor(acc, off, 32);
    if (L == 0) atomicAdd(out, acc * (1.0f / (64.0f * 64.0f)));
}

extern "C" void kernel_launch(void* const* d_in, const int* in_sizes, int n_in,
                              void* d_out, int out_size, void* d_ws, size_t ws_size,
                              hipStream_t stream) {
    const float* predsS = (const float*)d_in[0];
    const float* predsT = (const float*)d_in[1];
    const float* depth  = (const float*)d_in[2];
    float* out = (float*)d_out;

    char* ws = (char*)d_ws;
    uint8_t* idx   = (uint8_t*)(ws + OFF_IDX);
    float*   sumsS = (float*)(ws + OFF_SUMS_S);
    float*   sumsT = (float*)(ws + OFF_SUMS_T);
    float*   cnts  = (float*)(ws + OFF_CNT);
    float*   protS = (float*)(ws + OFF_PS);
    float*   protT = (float*)(ws + OFF_PT);

    k_zero  <<<(ZERO_FLOATS + 255) / 256, 256, 0, stream>>>(sumsS, out);
    k_idx   <<<P_ / (256 * 4),            256, 0, stream>>>(depth, idx, cnts);
    k_accum <<<2 * N_ * C_,               256, 0, stream>>>(predsS, predsT, idx, sumsS, sumsT);
    k_protos<<<NBINS,                     C_,  0, stream>>>(sumsS, cnts, protS);
    k_protos<<<NBINS,                     C_,  0, stream>>>(sumsT, cnts, protT);
    k_gram  <<<16,                        32,  0, stream>>>(protS, protT, out);
}